// cheb_AE_16037407883741
// MI455X (gfx1250) — compile-verified
//
#include <hip/hip_runtime.h>

typedef __attribute__((ext_vector_type(2))) float v2f;
typedef __attribute__((ext_vector_type(8))) float v8f;

namespace {
constexpr int  B_   = 8;
constexpr int  N0_  = 50000;
constexpr int  N1_  = 12500;
constexpr int  E_   = 400000;
constexpr int  FIN_ = 3;
constexpr int  C_   = 32;
constexpr int  K_   = 6;
constexpr int  Z_   = 128;
constexpr long MR_  = (long)B_ * N0_;   // 400000 fused (batch,vertex) rows
constexpr int  MT_  = (int)(MR_ / 16);  // 25000 16-row tiles
constexpr int  GX_  = MT_ / 8;          // 3125 blocks of 8 waves
constexpr int  SE_  = 4 * K_;           // 24: encoder Tall row stride (4-padded FIN)
constexpr int  SD_  = C_ * K_;          // 192: decoder Tall row stride
}

static inline int blks(long n, int t) { return (int)((n + t - 1) / t); }

// ---------------- elementwise helpers ----------------
__global__ void zero_kernel(float* __restrict__ p, long n) {
  long t = (long)blockIdx.x * blockDim.x + threadIdx.x;
  if (t < n) p[t] = 0.0f;
}

// column-slice ops on Tall with row stride S
__global__ void zero_cols_kernel(float* __restrict__ T, int S, int ofs, int chan) {
  long t = (long)blockIdx.x * blockDim.x + threadIdx.x;
  if (t >= MR_ * chan) return;
  long m = t / chan;
  int c = (int)(t % chan);
  T[m * S + ofs + c] = 0.0f;
}

__global__ void negcopy_cols_kernel(float* __restrict__ T, int S, int dofs, int sofs,
                                    int chan) {
  long t = (long)blockIdx.x * blockDim.x + threadIdx.x;
  if (t >= MR_ * chan) return;
  long m = t / chan;
  int c = (int)(t % chan);
  T[m * S + dofs + c] = -T[m * S + sofs + c];
}

__global__ void copyin_cols_kernel(float* __restrict__ T, int S, int dofs,
                                   const float* __restrict__ src, int chan) {
  long t = (long)blockIdx.x * blockDim.x + threadIdx.x;
  if (t >= MR_ * chan) return;
  long m = t / chan;
  int c = (int)(t % chan);
  T[m * S + dofs + c] = src[t];
}

// x [MR,3] -> TencAll cols 0..3 (pad col 3 = 0), row stride SE_
__global__ void pack_x_kernel(const float* __restrict__ x, float* __restrict__ T) {
  long t = (long)blockIdx.x * blockDim.x + threadIdx.x;
  if (t >= MR_ * 4) return;
  int c = (int)(t & 3);
  long m = t >> 2;
  T[m * SE_ + c] = (c < FIN_) ? x[m * FIN_ + c] : 0.0f;
}

// W_enc0 [6,3,32] -> [24,32], rows 4k+3 zeroed
__global__ void pack_wenc_kernel(const float* __restrict__ w, float* __restrict__ o) {
  int t = blockIdx.x * blockDim.x + threadIdx.x;
  if (t >= SE_ * C_) return;
  int col = t & 31, r = (t >> 5) & 3, k = t >> 7;
  o[t] = (r < FIN_) ? w[(k * FIN_ + r) * C_ + col] : 0.0f;
}

// W_dec1 [6,32,3] -> [192,16], cols 3..15 zeroed
__global__ void pack_wdec1_kernel(const float* __restrict__ w, float* __restrict__ o) {
  int t = blockIdx.x * blockDim.x + threadIdx.x;
  if (t >= SD_ * 16) return;
  int col = t & 15, row = t >> 4;  // row = k*32 + cin
  o[t] = (col < FIN_) ? w[row * FIN_ + col] : 0.0f;
}

// Cpad [MR,16] -> out [MR,3]
__global__ void unpad_out_kernel(const float* __restrict__ cp, float* __restrict__ o) {
  long t = (long)blockIdx.x * blockDim.x + threadIdx.x;
  if (t >= MR_ * FIN_) return;
  long m = t / FIN_;
  int c = (int)(t % FIN_);
  o[t] = cp[m * 16 + c];
}

__global__ void bias_relu_kernel(float* __restrict__ h, const float* __restrict__ bias,
                                 long n, int C) {
  long t = (long)blockIdx.x * blockDim.x + threadIdx.x;
  if (t >= n) return;
  float v = h[t] + bias[(int)(t % C)];
  h[t] = v > 0.0f ? v : 0.0f;
}

__global__ void init_z_kernel(float* __restrict__ z16, const float* __restrict__ bias) {
  int t = blockIdx.x * blockDim.x + threadIdx.x;
  if (t >= 16 * Z_) return;
  int r = t >> 7, j = t & 127;
  z16[t] = (r < B_) ? bias[j] : 0.0f;
}

// ------- graph propagate: Tall[:,outOfs..] += scale*norm[e]*Tall[src,inOfs..] ---------
__global__ void scatter_pad4_kernel(float* __restrict__ T, const int* __restrict__ src,
                                    const int* __restrict__ dst, const float* __restrict__ norm,
                                    float scale, int inOfs, int outOfs) {
  long t = (long)blockIdx.x * blockDim.x + threadIdx.x;
  if (t >= (long)E_ * B_) return;
  int e = (int)(t % E_);
  int b = (int)(t / E_);
  float w = scale * norm[e];
  const float* in = T + ((long)b * N0_ + src[e]) * SE_ + inOfs;
  float* o = T + ((long)b * N0_ + dst[e]) * SE_ + outOfs;
  atomicAdd(o + 0, w * in[0]);
  atomicAdd(o + 1, w * in[1]);
  atomicAdd(o + 2, w * in[2]);
}

__global__ void scatter32_kernel(float* __restrict__ T, const int* __restrict__ src,
                                 const int* __restrict__ dst, const float* __restrict__ norm,
                                 float scale, int inOfs, int outOfs) {
  long t = (long)blockIdx.x * blockDim.x + threadIdx.x;
  if (t >= (long)E_ * B_ * 8) return;
  int quad = (int)(t & 7);
  long r = t >> 3;
  int e = (int)(r % E_);
  int b = (int)(r / E_);
  float w = scale * norm[e];
  const float4 vi = *reinterpret_cast<const float4*>(
      T + ((long)b * N0_ + src[e]) * SD_ + inOfs + quad * 4);
  float* o = T + ((long)b * N0_ + dst[e]) * SD_ + outOfs + quad * 4;
  atomicAdd(o + 0, w * vi.x);
  atomicAdd(o + 1, w * vi.y);
  atomicAdd(o + 2, w * vi.z);
  atomicAdd(o + 3, w * vi.w);
}

// ---------------- surface pooling (gather-only; rows are repeat(arange)) ----------------
__global__ void pool_down_kernel(const float* __restrict__ h, const int* __restrict__ col,
                                 const float* __restrict__ val, float* __restrict__ o) {
  long t = (long)blockIdx.x * blockDim.x + threadIdx.x;
  if (t >= (long)B_ * N1_ * C_) return;
  int c = (int)(t % C_);
  long r = t / C_;
  int n = (int)(r % N1_);
  int b = (int)(r / N1_);
  float acc = 0.0f;
#pragma unroll
  for (int q = 0; q < 4; ++q) {
    int cc = col[n * 4 + q];
    acc += val[n * 4 + q] * h[((long)b * N0_ + cc) * C_ + c];
  }
  o[t] = acc;  // rows 0..7 of padded [16,400000] A-matrix
}

// writes T0 slice (cols 0..31) of decoder Tall (row stride SD_)
__global__ void pool_up_kernel(const float* __restrict__ h, const int* __restrict__ col,
                               const float* __restrict__ val, float* __restrict__ T) {
  long t = (long)blockIdx.x * blockDim.x + threadIdx.x;
  if (t >= MR_ * C_) return;
  int c = (int)(t % C_);
  long r = t / C_;
  int n = (int)(r % N0_);
  int b = (int)(r / N0_);
  float acc = 0.0f;
#pragma unroll
  for (int q = 0; q < 3; ++q) {
    int cc = col[n * 3 + q];
    acc += val[n * 3 + q] * h[((long)b * N1_ + cc) * C_ + c];
  }
  T[((long)b * N0_ + n) * SD_ + c] = acc;
}

// ------ write-only WMMA GEMM: C[MR_,NCOLS] = A[MR_,KDIM] @ B[KDIM,NCOLS] ------
// B tile staged in LDS (shared by the 8 waves of the block).
template <int KDIM, int NCOLS>
__global__ __launch_bounds__(256) void gemm_tile_kernel(const float* __restrict__ A,
                                                        const float* __restrict__ Bm,
                                                        float* __restrict__ Cm) {
  __shared__ float lbs[KDIM * 16];
  const int nbase = blockIdx.y * 16;
  for (int i = threadIdx.x; i < KDIM * 16; i += 256) {
    int r = i >> 4, col = i & 15;
    lbs[i] = Bm[r * NCOLS + nbase + col];
  }
  __syncthreads();

  const int wave = threadIdx.x >> 5;
  const int lane = threadIdx.x & 31;
  const int hi = lane >> 4;   // K/M half split per ISA VGPR layout
  const int l16 = lane & 15;
  const int mbase = (blockIdx.x * 8 + wave) * 16;
  const int arow = mbase + l16;
  const int crow = mbase + hi * 8;

  v8f c = {};
#pragma unroll 8
  for (int kk = 0; kk < KDIM; kk += 4) {
    const int ka = kk + hi * 2;
    v2f a = *reinterpret_cast<const v2f*>(A + (size_t)arow * KDIM + ka);
    v2f b;
    b.x = lbs[ka * 16 + l16];
    b.y = lbs[(ka + 1) * 16 + l16];
    c = __builtin_amdgcn_wmma_f32_16x16x4_f32(false, a, false, b, (short)0, c, false, false);
  }
#pragma unroll
  for (int v = 0; v < 8; ++v) Cm[(size_t)(crow + v) * NCOLS + nbase + l16] = c[v];
}

// ---- encoder latent: z[16,128] += A[16,400000] @ W[400000,128], split-K + atomics ----
__global__ __launch_bounds__(32) void dense_enc_kernel(const float* __restrict__ A,
                                                       const float* __restrict__ W,
                                                       float* __restrict__ z16, int kchunk) {
  const int lane = threadIdx.x & 31;
  const int hi = lane >> 4, l16 = lane & 15;
  const int bcol = blockIdx.x * 16 + l16;
  const long k0 = (long)blockIdx.y * kchunk;
  v8f c = {};
  for (long kk = k0; kk < k0 + kchunk; kk += 16) {
#pragma unroll
    for (int u = 0; u < 16; u += 4) {
      long ka = kk + u + hi * 2;
      v2f a = *reinterpret_cast<const v2f*>(A + (long)l16 * 400000 + ka);
      v2f b;
      b.x = W[ka * Z_ + bcol];
      b.y = W[(ka + 1) * Z_ + bcol];
      c = __builtin_amdgcn_wmma_f32_16x16x4_f32(false, a, false, b, (short)0, c, false, false);
    }
  }
#pragma unroll
  for (int v = 0; v < 8; ++v) {
    int row = hi * 8 + v;
    if (row < B_) atomicAdd(&z16[row * Z_ + bcol], c[v]);  // rows 8..15 are zero anyway
  }
}

// ---- decoder latent: h[8,400000] = relu(z[16,128] @ W[128,400000] + b) ----
__global__ __launch_bounds__(256) void dense_dec_kernel(const float* __restrict__ z16,
                                                        const float* __restrict__ W,
                                                        const float* __restrict__ bias,
                                                        float* __restrict__ o) {
  const int wave = threadIdx.x >> 5, lane = threadIdx.x & 31;
  const int hi = lane >> 4, l16 = lane & 15;
  const int col = (blockIdx.x * 8 + wave) * 16 + l16;
  v8f c = {};
#pragma unroll
  for (int kk = 0; kk < Z_; kk += 4) {
    int ka = kk + hi * 2;
    v2f a = *reinterpret_cast<const v2f*>(z16 + l16 * Z_ + ka);
    v2f b;
    b.x = W[(long)ka * 400000 + col];
    b.y = W[(long)(ka + 1) * 400000 + col];
    c = __builtin_amdgcn_wmma_f32_16x16x4_f32(false, a, false, b, (short)0, c, false, false);
  }
#pragma unroll
  for (int v = 0; v < 8; ++v) {
    int row = hi * 8 + v;
    if (row < B_) {
      float t = c[v] + bias[col];
      o[(long)row * 400000 + col] = t > 0.0f ? t : 0.0f;
    }
  }
}

// ---------------- host driver ----------------
extern "C" void kernel_launch(void* const* d_in, const int* in_sizes, int n_in, void* d_out,
                              int out_size, void* d_ws, size_t ws_size, hipStream_t stream) {
  (void)in_sizes; (void)n_in; (void)out_size; (void)ws_size;
  const float* x       = (const float*)d_in[0];
  const int*   edge    = (const int*)d_in[1];   // [2,E]: src=edge, dst=edge+E
  const float* Anorm   = (const float*)d_in[2];
  const int*   downIdx = (const int*)d_in[3];   // [2, N1*4]; cols at +N1*4
  const float* downVal = (const float*)d_in[4];
  const int*   upIdx   = (const int*)d_in[5];   // [2, N0*3]; cols at +N0*3
  const float* upVal   = (const float*)d_in[6];
  const float* Wenc0   = (const float*)d_in[7];   // [6,3,32]
  const float* benc0   = (const float*)d_in[8];
  const float* Wdec0   = (const float*)d_in[9];   // [6,32,32] == [192,32]
  const float* bdec0   = (const float*)d_in[10];
  const float* Wdec1   = (const float*)d_in[11];  // [6,32,3]
  const float* encLinW = (const float*)d_in[12];  // [400000,128]
  const float* encLinB = (const float*)d_in[13];
  const float* decLinW = (const float*)d_in[14];  // [128,400000]
  const float* decLinB = (const float*)d_in[15];
  float* out = (float*)d_out;

  // workspace layout (floats)
  float* w = (float*)d_ws;
  float* TdAll    = w;  w += MR_ * SD_;           // 76.8M floats; decoder concat-K buffer
  float* hMid     = w;  w += MR_ * C_;            // 12.8M
  float* Cpad     = w;  w += MR_ * 16;            // 6.4M
  float* hDownPad = w;  w += (long)16 * 400000;   // 6.4M; A for dense_enc, rows 8..15 zero
  float* z16      = w;  w += 16 * Z_;
  float* hDec     = w;  w += (long)B_ * N1_ * C_; // 3.2M
  float* WpEnc    = w;  w += SE_ * C_;            // [24,32]
  float* WpDec1   = w;  w += SD_ * 16;            // [192,16]
  // encoder-phase overlays inside TdAll (dead before decoder writes TdAll):
  float* TencAll  = TdAll;                        // [MR,24] = 9.6M floats
  float* hEnc     = TdAll + MR_ * SE_;            // 12.8M floats

  // ---- weight packing ----
  pack_wenc_kernel<<<blks(SE_ * C_, 256), 256, 0, stream>>>(Wenc0, WpEnc);
  pack_wdec1_kernel<<<blks(SD_ * 16, 256), 256, 0, stream>>>(Wdec1, WpDec1);

  const int* src = edge;
  const int* dst = edge + E_;
  auto scat4 = [&](float s, int iofs, int oofs) {
    scatter_pad4_kernel<<<blks((long)E_ * B_, 256), 256, 0, stream>>>(TencAll, src, dst, Anorm,
                                                                      s, iofs, oofs);
  };
  auto scat32 = [&](float s, int iofs, int oofs) {
    scatter32_kernel<<<blks((long)E_ * B_ * 8, 256), 256, 0, stream>>>(TdAll, src, dst, Anorm,
                                                                       s, iofs, oofs);
  };
  // decoder Chebyshev recursion chain into TdAll slices (T0 already in cols 0..31)
  auto dec_chain = [&]() {
    zero_cols_kernel<<<blks(MR_ * C_, 256), 256, 0, stream>>>(TdAll, SD_, C_, C_);
    scat32(1.0f, 0, C_);
    for (int k = 2; k < K_; ++k) {
      negcopy_cols_kernel<<<blks(MR_ * C_, 256), 256, 0, stream>>>(TdAll, SD_, k * C_,
                                                                   (k - 2) * C_, C_);
      scat32(2.0f, (k - 1) * C_, k * C_);
    }
  };

  // ---------------- encoder ----------------
  pack_x_kernel<<<blks(MR_ * 4, 256), 256, 0, stream>>>(x, TencAll);
  zero_cols_kernel<<<blks(MR_ * 4, 256), 256, 0, stream>>>(TencAll, SE_, 4, 4);
  scat4(1.0f, 0, 4);
  for (int k = 2; k < K_; ++k) {
    negcopy_cols_kernel<<<blks(MR_ * 4, 256), 256, 0, stream>>>(TencAll, SE_, k * 4,
                                                                (k - 2) * 4, 4);
    scat4(2.0f, (k - 1) * 4, k * 4);
  }
  gemm_tile_kernel<SE_, C_><<<dim3(GX_, C_ / 16), 256, 0, stream>>>(TencAll, WpEnc, hEnc);
  bias_relu_kernel<<<blks(MR_ * C_, 256), 256, 0, stream>>>(hEnc, benc0, MR_ * C_, C_);
  pool_down_kernel<<<blks((long)B_ * N1_ * C_, 256), 256, 0, stream>>>(
      hEnc, downIdx + (long)N1_ * 4, downVal, hDownPad);
  zero_kernel<<<blks((long)8 * 400000, 256), 256, 0, stream>>>(hDownPad + (long)8 * 400000,
                                                               (long)8 * 400000);
  init_z_kernel<<<blks(16 * Z_, 256), 256, 0, stream>>>(z16, encLinB);
  dense_enc_kernel<<<dim3(Z_ / 16, 250), 32, 0, stream>>>(hDownPad, encLinW, z16, 1600);

  // ---------------- decoder ----------------
  dense_dec_kernel<<<GX_, 256, 0, stream>>>(z16, decLinW, decLinB, hDec);
  pool_up_kernel<<<blks(MR_ * C_, 256), 256, 0, stream>>>(hDec, upIdx + (long)N0_ * 3, upVal,
                                                          TdAll);
  dec_chain();
  gemm_tile_kernel<SD_, C_><<<dim3(GX_, C_ / 16), 256, 0, stream>>>(TdAll, Wdec0, hMid);
  bias_relu_kernel<<<blks(MR_ * C_, 256), 256, 0, stream>>>(hMid, bdec0, MR_ * C_, C_);
  copyin_cols_kernel<<<blks(MR_ * C_, 256), 256, 0, stream>>>(TdAll, SD_, 0, hMid, C_);
  dec_chain();
  gemm_tile_kernel<SD_, 16><<<dim3(GX_, 1), 256, 0, stream>>>(TdAll, WpDec1, Cpad);
  unpad_out_kernel<<<blks(MR_ * FIN_, 256), 256, 0, stream>>>(Cpad, out);
}